// RGCN_7851200217493
// MI455X (gfx1250) — compile-verified
//
#include <hip/hip_runtime.h>
#include <hip/hip_bf16.h>

typedef float v2f __attribute__((ext_vector_type(2)));
typedef float v8f __attribute__((ext_vector_type(8)));

#define DDIM 256
#define LDSTRIDE 260   // 260 % 64 = 4 -> 16 consecutive rows hit distinct banks
#define MTILES 5
#define MROWS (MTILES * 16)   // 80 rows per block; 20000 / 80 = 250 blocks exact

// ---------------------------------------------------------------------------
// Fused triple GEMM: C0 = A@B0, C1 = A@B1, C2 = A@B2  (A [M,256], B [256,256])
// One block = 80-row M strip staged in LDS; 16 waves, each owns a 16-col N
// tile and drives 15 accumulator chains (3 B-matrices x 5 M-tiles), so each
// B fragment load feeds 5 WMMAs and each A fragment feeds 3.
// ---------------------------------------------------------------------------
__global__ __launch_bounds__(512)
void rgcn_gemm3_wmma(const float* __restrict__ A,
                     const float* __restrict__ B0, const float* __restrict__ B1,
                     const float* __restrict__ B2,
                     float* __restrict__ C0, float* __restrict__ C1,
                     float* __restrict__ C2) {
    __shared__ float As[MROWS * LDSTRIDE];   // 83.2 KB (<= 320 KB/WGP on CDNA5)
    const int m0 = blockIdx.x * MROWS;

    for (int i = threadIdx.x; i < MROWS * DDIM; i += 512) {
        int r = i >> 8, c = i & 255;
        As[r * LDSTRIDE + c] = A[(size_t)(m0 + r) * DDIM + c];
    }
    __syncthreads();

    const int lane  = threadIdx.x & 31;
    const int wave  = threadIdx.x >> 5;       // 0..15
    const int mrow  = lane & 15;              // A fragment row within tile
    const int khalf = (lane >> 4) * 2;        // lanes 16-31 hold K+2,K+3
    const int ncol  = wave * 16 + (lane & 15);

    v8f acc[3][MTILES];
#pragma unroll
    for (int b = 0; b < 3; ++b)
#pragma unroll
        for (int t = 0; t < MTILES; ++t)
            acc[b][t] = (v8f){0.f,0.f,0.f,0.f,0.f,0.f,0.f,0.f};

#pragma unroll 2
    for (int k0 = 0; k0 < DDIM; k0 += 4) {
        size_t r0 = (size_t)(k0 + khalf) * DDIM + ncol;
        size_t r1 = r0 + DDIM;
        v2f b0, b1, b2;
        b0.x = B0[r0]; b0.y = B0[r1];
        b1.x = B1[r0]; b1.y = B1[r1];
        b2.x = B2[r0]; b2.y = B2[r1];
#pragma unroll
        for (int t = 0; t < MTILES; ++t) {
            v2f a;
            a.x = As[(t * 16 + mrow) * LDSTRIDE + k0 + khalf + 0];
            a.y = As[(t * 16 + mrow) * LDSTRIDE + k0 + khalf + 1];
            acc[0][t] = __builtin_amdgcn_wmma_f32_16x16x4_f32(false, a, false, b0,
                                                              (short)0, acc[0][t], false, false);
            acc[1][t] = __builtin_amdgcn_wmma_f32_16x16x4_f32(false, a, false, b1,
                                                              (short)0, acc[1][t], false, false);
            acc[2][t] = __builtin_amdgcn_wmma_f32_16x16x4_f32(false, a, false, b2,
                                                              (short)0, acc[2][t], false, false);
        }
    }

    const int rsub = (lane >> 4) * 8;         // D: VGPR j -> row tile*16 + rsub + j
#pragma unroll
    for (int t = 0; t < MTILES; ++t) {
#pragma unroll
        for (int j = 0; j < 8; ++j) {
            size_t o = (size_t)(m0 + t * 16 + rsub + j) * DDIM + ncol;
            C0[o] = acc[0][t][j];
            C1[o] = acc[1][t][j];
            C2[o] = acc[2][t][j];
        }
    }
}

// ---------------------------------------------------------------------------
// CSR-by-dst build (once per call; graph is layer-invariant)
// ---------------------------------------------------------------------------
__global__ __launch_bounds__(256)
void rgcn_zero_i32(int* __restrict__ p, int n) {
    int i = blockIdx.x * 256 + threadIdx.x;
    if (i < n) p[i] = 0;
}

__global__ __launch_bounds__(256)
void rgcn_deg_hist(const int* __restrict__ dst, int* __restrict__ deg, int E) {
    int e = blockIdx.x * 256 + threadIdx.x;
    if (e < E) atomicAdd(&deg[dst[e]], 1);
}

// Single-block exclusive scan over N degrees -> rowstart[N+1]
__global__ __launch_bounds__(1024)
void rgcn_scan(const int* __restrict__ deg, int* __restrict__ rowstart, int N) {
    __shared__ int partial[1024];
    const int t = threadIdx.x;
    const int CH = (N + 1023) / 1024;
    const int base = t * CH;

    int local = 0;
    for (int i = 0; i < CH; ++i) {
        int idx = base + i;
        if (idx < N) local += deg[idx];
    }
    partial[t] = local;
    __syncthreads();
    for (int off = 1; off < 1024; off <<= 1) {   // Hillis-Steele inclusive scan
        int v = (t >= off) ? partial[t - off] : 0;
        __syncthreads();
        partial[t] += v;
        __syncthreads();
    }
    int prefix = (t == 0) ? 0 : partial[t - 1];
    for (int i = 0; i < CH; ++i) {
        int idx = base + i;
        if (idx < N) { rowstart[idx] = prefix; prefix += deg[idx]; }
    }
    if (t == 0) rowstart[N] = partial[1023];
}

// adj[rowstart[d] + cursor[d]++] = (src << 1) | etype
__global__ __launch_bounds__(256)
void rgcn_csr_fill(const int* __restrict__ src, const int* __restrict__ dst,
                   const int* __restrict__ etype, const int* __restrict__ rowstart,
                   int* __restrict__ cursor, int* __restrict__ adj, int E) {
    int e = blockIdx.x * 256 + threadIdx.x;
    if (e >= E) return;
    int d = dst[e];
    int pos = atomicAdd(&cursor[d], 1);
    adj[rowstart[d] + pos] = (src[e] << 1) | (etype[e] & 1);
}

// ---------------------------------------------------------------------------
// Fused gather-aggregate + LayerNorm + bias + self-loop + optional ReLU.
// One wave per node; node row in registers (lane owns cols [lane*8, lane*8+8)
// -> b128 loads/stores), wave32 shuffle reductions for mean/var.
// ---------------------------------------------------------------------------
__global__ __launch_bounds__(256)
void rgcn_aggregate_finalize(const float* __restrict__ y0, const float* __restrict__ y1,
                             const int* __restrict__ rowstart, const int* __restrict__ adj,
                             const float* __restrict__ selfv,
                             const float* __restrict__ gamma, const float* __restrict__ beta,
                             const float* __restrict__ bias, float* __restrict__ out,
                             int N, int doRelu) {
    int node = blockIdx.x * 8 + (threadIdx.x >> 5);
    if (node >= N) return;
    int lane = threadIdx.x & 31;
    const int cbase = lane * 8;

    float v[8];
#pragma unroll
    for (int j = 0; j < 8; ++j) v[j] = 0.f;

    const int beg = rowstart[node], end = rowstart[node + 1];
    for (int i = beg; i < end; ++i) {
        int packed = adj[i];                  // wave-uniform -> scalar load
        const float* row = ((packed & 1) ? y1 : y0) + (size_t)(packed >> 1) * DDIM + cbase;
#pragma unroll
        for (int j = 0; j < 8; ++j) v[j] += row[j];
    }

    float s = 0.f;
#pragma unroll
    for (int j = 0; j < 8; ++j) s += v[j];
#pragma unroll
    for (int m = 16; m >= 1; m >>= 1) s += __shfl_xor(s, m, 32);
    float mu = s * (1.f / 256.f);

    float q = 0.f;
#pragma unroll
    for (int j = 0; j < 8; ++j) { float t = v[j] - mu; q += t * t; }
#pragma unroll
    for (int m = 16; m >= 1; m >>= 1) q += __shfl_xor(q, m, 32);
    float rstd = rsqrtf(q * (1.f / 256.f) + 1e-5f);

    const float* srow = selfv + (size_t)node * DDIM + cbase;
    float* orow = out + (size_t)node * DDIM + cbase;
#pragma unroll
    for (int j = 0; j < 8; ++j) {
        int c = cbase + j;
        float hn = (v[j] - mu) * rstd * gamma[c] + beta[c] + bias[c] + srow[j];
        orow[j] = doRelu ? fmaxf(hn, 0.f) : hn;
    }
}

// ---------------------------------------------------------------------------
extern "C" void kernel_launch(void* const* d_in, const int* in_sizes, int n_in,
                              void* d_out, int out_size, void* d_ws, size_t ws_size,
                              hipStream_t stream) {
    const float* x      = (const float*)d_in[0];  // [N, 256]
    const float* W      = (const float*)d_in[1];  // [3, 2, 256, 256]
    const float* bias   = (const float*)d_in[2];  // [3, 256]
    const float* loop_w = (const float*)d_in[3];  // [3, 256, 256]
    const float* gamma  = (const float*)d_in[4];  // [3, 256]
    const float* beta   = (const float*)d_in[5];  // [3, 256]
    const int*   src    = (const int*)d_in[6];    // [E]
    const int*   dst    = (const int*)d_in[7];    // [E]
    const int*   etype  = (const int*)d_in[8];    // [E]
    float* out = (float*)d_out;

    const int N = 20000, E = 320000;
    const size_t BUF = (size_t)N * DDIM;          // floats per buffer

    float* ws  = (float*)d_ws;
    float* h_a = ws;
    float* h_b = ws + BUF;
    float* y0  = ws + 2 * BUF;
    float* y1  = ws + 3 * BUF;
    float* sv  = ws + 4 * BUF;
    int* ibase     = (int*)(ws + 5 * BUF);
    int* deg       = ibase;                       // [N]
    int* cursor    = ibase + N;                   // [N]
    int* rowstart  = ibase + 2 * N;               // [N+1]
    int* adj       = ibase + 3 * N + 1;           // [E]

    // ---- Build CSR by destination (once) ----
    rgcn_zero_i32<<<(2 * N + 255) / 256, 256, 0, stream>>>(deg, 2 * N); // deg+cursor
    rgcn_deg_hist<<<(E + 255) / 256, 256, 0, stream>>>(dst, deg, E);
    rgcn_scan<<<1, 1024, 0, stream>>>(deg, rowstart, N);
    rgcn_csr_fill<<<(E + 255) / 256, 256, 0, stream>>>(src, dst, etype,
                                                       rowstart, cursor, adj, E);

    // ---- Layers ----
    const float* h = x;
    for (int l = 0; l < 3; ++l) {
        const float* W0 = W + ((size_t)l * 2 + 0) * DDIM * DDIM;
        const float* W1 = W + ((size_t)l * 2 + 1) * DDIM * DDIM;
        const float* Lw = loop_w + (size_t)l * DDIM * DDIM;

        rgcn_gemm3_wmma<<<N / MROWS, 512, 0, stream>>>(h, W0, W1, Lw, y0, y1, sv);

        float* hn = (l == 2) ? out : (l == 0 ? h_a : h_b);
        rgcn_aggregate_finalize<<<(N + 7) / 8, 256, 0, stream>>>(
            y0, y1, rowstart, adj, sv,
            gamma + (size_t)l * DDIM, beta + (size_t)l * DDIM,
            bias + (size_t)l * DDIM, hn, N, (l < 2) ? 1 : 0);
        h = hn;
    }
}